// GraphConvolution_14276471292058
// MI455X (gfx1250) — compile-verified
//
#include <hip/hip_runtime.h>
#include <hip/hip_bf16.h>

typedef __attribute__((ext_vector_type(2))) float v2f;
typedef __attribute__((ext_vector_type(8))) float v8f;

#define D_IN  256
#define D_OUT 256

// ---------------------------------------------------------------------------
// Kernel 0: repack W into B-fragment-native layout.
// Wp2[p*256 + c] = (W[2p][c], W[2p+1][c])  for p in [0,128), c in [0,256).
// A B-fragment for K-step k, lane-half h, column c is then the single float2
// at Wp2[(k/2 + h)*256 + c]  -> one global_load_b64 per fragment.
// ---------------------------------------------------------------------------
__global__ __launch_bounds__(256) void gcn_repack_w(
    const float* __restrict__ W, float2* __restrict__ Wp2) {
  int i = blockIdx.x * blockDim.x + threadIdx.x;   // i in [0, 128*256)
  int p = i >> 8;
  int c = i & 255;
  float2 v;
  v.x = W[(2 * p) * D_OUT + c];
  v.y = W[(2 * p + 1) * D_OUT + c];
  Wp2[i] = v;
}

// ---------------------------------------------------------------------------
// Kernel A: XW = X @ W   (fp32 WMMA 16x16x4)
// Block = 256 threads = 8 waves. Each wave computes a 32x64 strip:
// 2 row-strips of 16 x 4 col-tiles of 16 -> 8 accumulators (64 VGPRs).
// Per K=4 step: 2 A b64 loads + 4 B b64 loads feed 8 WMMAs.
// ---------------------------------------------------------------------------
__global__ __launch_bounds__(256) void gcn_gemm_wmma(
    const float* __restrict__ X, const float2* __restrict__ Wp2,
    float* __restrict__ XW, int n_rows) {
  const int lane   = threadIdx.x & 31;
  const int waveId = threadIdx.x >> 5;
  const int half   = lane >> 4;   // 0: lanes 0-15, 1: lanes 16-31
  const int idx    = lane & 15;

  const int row0 = blockIdx.y * 256 + waveId * 32;   // 32 rows per wave
  const int col0 = blockIdx.x * 64;

  v8f acc[2][4] = {};

  // Clamp A rows so loads stay in-bounds; stores are predicated instead
  // (WMMA requires EXEC == all ones, so no divergence before the MMA loop).
  int arow0 = row0 + idx;       if (arow0 >= n_rows) arow0 = n_rows - 1;
  int arow1 = row0 + 16 + idx;  if (arow1 >= n_rows) arow1 = n_rows - 1;
  const float* __restrict__ ap0 = X + (size_t)arow0 * D_IN;
  const float* __restrict__ ap1 = X + (size_t)arow1 * D_IN;

  const float2* __restrict__ bcol = Wp2 + col0 + idx;   // + (k/2+half)*256 + 16*t

  #pragma unroll 4
  for (int k = 0; k < D_IN; k += 4) {
    // A fragments: VGPR j holds A[m][k + 2*half + j] -> contiguous float2
    v2f a0 = *(const v2f*)(ap0 + k + 2 * half);
    v2f a1 = *(const v2f*)(ap1 + k + 2 * half);

    // B fragments: one b64 each from the repacked layout
    const float2* __restrict__ bb = bcol + (size_t)(k / 2 + half) * D_OUT;
    v2f b[4];
    #pragma unroll
    for (int t = 0; t < 4; ++t) {
      float2 bv = bb[16 * t];
      b[t].x = bv.x; b[t].y = bv.y;
    }

    #pragma unroll
    for (int t = 0; t < 4; ++t) {
      acc[0][t] = __builtin_amdgcn_wmma_f32_16x16x4_f32(
          false, a0, false, b[t], (short)0, acc[0][t], false, false);
      acc[1][t] = __builtin_amdgcn_wmma_f32_16x16x4_f32(
          false, a1, false, b[t], (short)0, acc[1][t], false, false);
    }
  }

  // C/D layout: VGPR i holds row (strip_row0 + i + 8*half), col (col0 + idx)
  const int ccol = col0 + idx;
  #pragma unroll
  for (int s = 0; s < 2; ++s) {
    #pragma unroll
    for (int i = 0; i < 8; ++i) {
      const int r = row0 + 16 * s + i + 8 * half;
      if (r < n_rows) {
        float* __restrict__ zp = XW + (size_t)r * D_OUT + ccol;
        zp[0]  = acc[s][0][i];
        zp[16] = acc[s][1][i];
        zp[32] = acc[s][2][i];
        zp[48] = acc[s][3][i];
      }
    }
  }
}

// ---------------------------------------------------------------------------
// Kernel B: Z[i][d] = bias[d]   (d_out is poisoned; must initialize)
// ---------------------------------------------------------------------------
__global__ __launch_bounds__(256) void gcn_init_bias(
    const float* __restrict__ bias, float* __restrict__ Z, int total) {
  int i = blockIdx.x * blockDim.x + threadIdx.x;
  if (i < total) Z[i] = bias[i & (D_OUT - 1)];
}

// ---------------------------------------------------------------------------
// Kernel C: Z[dst] += w_e * XW[src].  One wave per edge.
// Each lane handles 2 float4 (8 floats) of the 256-float row.
// ---------------------------------------------------------------------------
__global__ __launch_bounds__(256) void gcn_edge_scatter(
    const float* __restrict__ XW, const int* __restrict__ edge_src,
    const int* __restrict__ edge_dst, const float* __restrict__ edge_w,
    float* __restrict__ Z, int n_edges) {
  const int wave = (int)((blockIdx.x * (unsigned)blockDim.x + threadIdx.x) >> 5);
  const int lane = threadIdx.x & 31;
  if (wave >= n_edges) return;

  const int   s = edge_src[wave];
  const int   d = edge_dst[wave];
  const float w = edge_w[wave];

  const float4* __restrict__ xs = (const float4*)(XW + (size_t)s * D_OUT);
  float* __restrict__ zp = Z + (size_t)d * D_OUT;

  #pragma unroll
  for (int t = 0; t < 2; ++t) {
    const int v4 = lane + 32 * t;          // float4 index within the row
    float4 v = xs[v4];
    float* p = zp + v4 * 4;
    atomicAdd(p + 0, w * v.x);
    atomicAdd(p + 1, w * v.y);
    atomicAdd(p + 2, w * v.z);
    atomicAdd(p + 3, w * v.w);
  }
}

// ---------------------------------------------------------------------------
extern "C" void kernel_launch(void* const* d_in, const int* in_sizes, int n_in,
                              void* d_out, int out_size, void* d_ws, size_t ws_size,
                              hipStream_t stream) {
  const float* x           = (const float*)d_in[0];   // [N, 256]
  const float* weight      = (const float*)d_in[1];   // [256, 256]
  const float* bias        = (const float*)d_in[2];   // [256]
  const int*   edge_src    = (const int*)d_in[3];     // [E]
  const int*   edge_dst    = (const int*)d_in[4];     // [E]
  const float* edge_weight = (const float*)d_in[5];   // [E]
  float*       z           = (float*)d_out;           // [N, 256]

  const int n_nodes = in_sizes[0] / D_IN;
  const int n_edges = in_sizes[3];

  float*  xw  = (float*)d_ws;                                  // [N, 256]
  float2* wp2 = (float2*)(xw + (size_t)n_nodes * D_OUT);       // [128, 256] float2

  // 0) Repack W for single-b64 B-fragment loads (32768 threads)
  gcn_repack_w<<<(128 * 256) / 256, 256, 0, stream>>>(weight, wp2);

  // 1) XW = X @ W  (WMMA fp32): 8 waves/block, 256 rows x 64 cols per block
  {
    dim3 grid(D_OUT / 64, (n_nodes + 255) / 256);
    gcn_gemm_wmma<<<grid, 256, 0, stream>>>(x, wp2, xw, n_nodes);
  }

  // 2) Z = bias (broadcast)
  {
    const int total = n_nodes * D_OUT;
    gcn_init_bias<<<(total + 255) / 256, 256, 0, stream>>>(bias, z, total);
  }

  // 3) Z[dst] += w_e * XW[src]
  {
    const long long waves  = (long long)n_edges;
    const long long blocks = (waves * 32 + 255) / 256;
    gcn_edge_scatter<<<(int)blocks, 256, 0, stream>>>(xw, edge_src, edge_dst,
                                                      edge_weight, z, n_edges);
  }
}